// NIC_52630529245770
// MI455X (gfx1250) — compile-verified
//
#include <hip/hip_runtime.h>

#define V_TOK 3000
#define H_DIM 1000
#define B_SZ  64
#define T_STEPS 20
#define G4 (4 * V_TOK)   // 12000 gate columns

typedef float v2f __attribute__((ext_vector_type(2)));
typedef float v8f __attribute__((ext_vector_type(8)));

// ---------------------------------------------------------------------------
// Prep: bsum = b_ih + b_hh ; argmax buffer = 0 (out0 is one-hot at col 0) ;
// outputs[:,0,:] = one-hot(START_ID=0).
// ---------------------------------------------------------------------------
__global__ void __launch_bounds__(256)
prep_kernel(const float* __restrict__ b_ih, const float* __restrict__ b_hh,
            float* __restrict__ bsum, int* __restrict__ amax,
            float* __restrict__ outputs) {
  int idx = blockIdx.x * blockDim.x + threadIdx.x;
  if (idx < G4) bsum[idx] = b_ih[idx] + b_hh[idx];
  if (idx < B_SZ) amax[idx] = 0;
  if (idx < B_SZ * V_TOK) {
    int b = idx / V_TOK;
    int v = idx - b * V_TOK;
    outputs[(size_t)b * T_STEPS * V_TOK + v] = (v == 0) ? 1.0f : 0.0f;
  }
}

// ---------------------------------------------------------------------------
// Gather: feed[b,:] = emb[ mask ? argmax_prev[b] : target[b, t-1] ]
// ---------------------------------------------------------------------------
__global__ void __launch_bounds__(256)
gather_kernel(const int* __restrict__ target, const int* __restrict__ sched_mask,
              const int* __restrict__ amax, const float* __restrict__ emb,
              float* __restrict__ feed, int tcol) {
  int b = blockIdx.x;
  int m = sched_mask[tcol];
  int row = (m > 0) ? amax[b] : target[b * T_STEPS + tcol];
  const float* src = emb + (size_t)row * H_DIM;
  float* dst = feed + (size_t)b * H_DIM;
  for (int j = threadIdx.x; j < H_DIM; j += blockDim.x) dst[j] = src[j];
}

// ---------------------------------------------------------------------------
// GEMM: gates[64,12000] = feed[64,1000] @ W_ih[12000,1000]^T + bsum
// Exact fp32 via V_WMMA_F32_16X16X4_F32.
// Each wave: one 16-row M tile x two 16-col N tiles (A frag reused -> inner
// loop is 3 loads : 2 WMMAs). Block = 4 waves covering all 64 M rows for one
// 32-col N strip (B lines shared across waves via WGP$). Grid = 12000/32.
// Fragment layout (ISA 7.12.2, 32-bit):
//   A 16x4 : lane<16 -> M=lane, vgpr{0,1} = K{0,1};  lane>=16 -> K{2,3}
//   B 4x16 : lane<16 -> N=lane, vgpr{0,1} = K{0,1};  lane>=16 -> K{2,3}
//   C 16x16: vgpr r, lane<16 -> (M=r, N=lane); lane>=16 -> (M=r+8, N=lane-16)
// ---------------------------------------------------------------------------
__global__ void __launch_bounds__(128)
gemm_kernel(const float* __restrict__ feed, const float* __restrict__ W,
            const float* __restrict__ bsum, float* __restrict__ gates) {
  int lane = threadIdx.x & 31;
  int wave = threadIdx.x >> 5;
  int n0 = blockIdx.x * 32;
  int m0 = wave * 16;
  int l15 = lane & 15;
  int hi  = lane >> 4;            // 0: K pair {0,1}, 1: K pair {2,3}

  const float* arow  = feed + (size_t)(m0 + l15) * H_DIM + 2 * hi;
  const float* brow0 = W    + (size_t)(n0 + l15) * H_DIM + 2 * hi;
  const float* brow1 = brow0 + (size_t)16 * H_DIM;

  v8f c0 = {0.f, 0.f, 0.f, 0.f, 0.f, 0.f, 0.f, 0.f};
  v8f c1 = {0.f, 0.f, 0.f, 0.f, 0.f, 0.f, 0.f, 0.f};
#pragma unroll 5
  for (int k = 0; k < H_DIM; k += 4) {
    v2f a  = *(const v2f*)(arow + k);
    v2f b0 = *(const v2f*)(brow0 + k);
    v2f b1 = *(const v2f*)(brow1 + k);
    c0 = __builtin_amdgcn_wmma_f32_16x16x4_f32(
        false, a, false, b0, (short)0, c0, false, false);
    c1 = __builtin_amdgcn_wmma_f32_16x16x4_f32(
        false, a, false, b1, (short)0, c1, false, false);
  }

  int na = n0 + l15;
  int nb = na + 16;
  float bva = bsum[na];
  float bvb = bsum[nb];
  int mb = m0 + hi * 8;
#pragma unroll
  for (int r = 0; r < 8; ++r) {
    gates[(size_t)(mb + r) * G4 + na] = c0[r] + bva;
    gates[(size_t)(mb + r) * G4 + nb] = c1[r] + bvb;
  }
}

// ---------------------------------------------------------------------------
// Activation + softmax + argmax, one block per batch row b.
//   h = sigmoid(o) * tanh(sigmoid(i) * tanh(g));  probs = softmax(h)
//   argmax(probs) == argmax(h) (first-index tie break, matching jnp.argmax)
// ---------------------------------------------------------------------------
__global__ void __launch_bounds__(256)
act_softmax_kernel(const float* __restrict__ gates, float* __restrict__ hbuf,
                   int* __restrict__ amax, int t, float* __restrict__ outputs) {
  __shared__ float sred[256];
  __shared__ int   sidx[256];
  int b = blockIdx.x;
  int tid = threadIdx.x;
  const float* gi = gates + (size_t)b * G4;             // i gate
  const float* gg = gi + 2 * V_TOK;                     // g gate
  const float* go = gi + 3 * V_TOK;                     // o gate
  float* hrow = hbuf + (size_t)b * V_TOK;

  float bmax = -3.4e38f;
  int bidx = 0;
  for (int v = tid; v < V_TOK; v += 256) {
    float i = gi[v], g = gg[v], o = go[v];
    float cc = (1.0f / (1.0f + expf(-i))) * tanhf(g);
    float h  = (1.0f / (1.0f + expf(-o))) * tanhf(cc);
    hrow[v] = h;
    if (h > bmax) { bmax = h; bidx = v; }               // strict > keeps first
  }
  sred[tid] = bmax; sidx[tid] = bidx;
  __syncthreads();
  for (int s = 128; s > 0; s >>= 1) {
    if (tid < s) {
      float om = sred[tid + s]; int oi = sidx[tid + s];
      if (om > sred[tid] || (om == sred[tid] && oi < sidx[tid])) {
        sred[tid] = om; sidx[tid] = oi;
      }
    }
    __syncthreads();
  }
  float rmax = sred[0];
  int   ridx = sidx[0];
  __syncthreads();

  float sum = 0.f;
  for (int v = tid; v < V_TOK; v += 256) sum += expf(hrow[v] - rmax);
  sred[tid] = sum;
  __syncthreads();
  for (int s = 128; s > 0; s >>= 1) {
    if (tid < s) sred[tid] += sred[tid + s];
    __syncthreads();
  }
  float inv = 1.0f / sred[0];

  float* orow = outputs + (size_t)b * T_STEPS * V_TOK + (size_t)t * V_TOK;
  for (int v = tid; v < V_TOK; v += 256)
    orow[v] = expf(hrow[v] - rmax) * inv;
  if (tid == 0) amax[b] = ridx;
}

// ---------------------------------------------------------------------------
// NLL per (b,t) row: log_softmax of the (already-softmaxed) probs, gather tgt.
// ---------------------------------------------------------------------------
__global__ void __launch_bounds__(256)
nll_kernel(const float* __restrict__ outputs, const int* __restrict__ target,
           float* __restrict__ nll) {
  __shared__ float sred[256];
  int row = blockIdx.x;                 // == b*T + t (flat [B,T] order)
  int tid = threadIdx.x;
  const float* p = outputs + (size_t)row * V_TOK;

  float m = -3.4e38f;
  for (int v = tid; v < V_TOK; v += 256) m = fmaxf(m, p[v]);
  sred[tid] = m;
  __syncthreads();
  for (int s = 128; s > 0; s >>= 1) {
    if (tid < s) sred[tid] = fmaxf(sred[tid], sred[tid + s]);
    __syncthreads();
  }
  float rmax = sred[0];
  __syncthreads();

  float sum = 0.f;
  for (int v = tid; v < V_TOK; v += 256) sum += expf(p[v] - rmax);
  sred[tid] = sum;
  __syncthreads();
  for (int s = 128; s > 0; s >>= 1) {
    if (tid < s) sred[tid] += sred[tid + s];
    __syncthreads();
  }
  if (tid == 0) {
    float lse = rmax + logf(sred[0]);
    nll[row] = lse - p[target[row]];
  }
}

// Deterministic final reduction (no float atomics -> replay-stable).
__global__ void __launch_bounds__(256)
loss_kernel(const float* __restrict__ nll, float* __restrict__ loss) {
  __shared__ float sred[256];
  int tid = threadIdx.x;
  float s = 0.f;
  for (int r = tid; r < B_SZ * T_STEPS; r += 256) s += nll[r];
  sred[tid] = s;
  __syncthreads();
  for (int st = 128; st > 0; st >>= 1) {
    if (tid < st) sred[tid] += sred[tid + st];
    __syncthreads();
  }
  if (tid == 0) loss[0] = sred[0] / (float)(B_SZ * T_STEPS);
}

// ---------------------------------------------------------------------------
extern "C" void kernel_launch(void* const* d_in, const int* in_sizes, int n_in,
                              void* d_out, int out_size, void* d_ws, size_t ws_size,
                              hipStream_t stream) {
  (void)in_sizes; (void)n_in; (void)out_size; (void)ws_size;
  // setup_inputs order: 0 image_features(unused) 1 target 2 sched_mask
  // 3 fc_w(unused) 4 fc_b(unused) 5 emb 6 W_ih 7 W_hh(unused: h=c=0 bug)
  // 8 b_ih 9 b_hh
  const int*   target = (const int*)d_in[1];
  const int*   sched  = (const int*)d_in[2];
  const float* emb    = (const float*)d_in[5];
  const float* W_ih   = (const float*)d_in[6];
  const float* b_ih   = (const float*)d_in[8];
  const float* b_hh   = (const float*)d_in[9];

  float* loss    = (float*)d_out;
  float* outputs = (float*)d_out + 1;   // [B, T, V] flat

  char* ws = (char*)d_ws;
  size_t off = 0;
  auto carve = [&](size_t bytes) -> void* {
    void* p = ws + off;
    off += (bytes + 255) & ~(size_t)255;
    return p;
  };
  float* feed  = (float*)carve((size_t)B_SZ * H_DIM * sizeof(float));   // 256 KB
  float* gates = (float*)carve((size_t)B_SZ * G4 * sizeof(float));      // 3.0 MB
  float* hbuf  = (float*)carve((size_t)B_SZ * V_TOK * sizeof(float));   // 768 KB
  float* bsum  = (float*)carve((size_t)G4 * sizeof(float));
  int*   amax  = (int*)carve((size_t)B_SZ * sizeof(int));
  float* nll   = (float*)carve((size_t)B_SZ * T_STEPS * sizeof(float));

  prep_kernel<<<(B_SZ * V_TOK + 255) / 256, 256, 0, stream>>>(
      b_ih, b_hh, bsum, amax, outputs);

  for (int t = 1; t < T_STEPS; ++t) {
    gather_kernel<<<B_SZ, 256, 0, stream>>>(target, sched, amax, emb, feed, t - 1);
    gemm_kernel<<<G4 / 32, 128, 0, stream>>>(feed, W_ih, bsum, gates);
    act_softmax_kernel<<<B_SZ, 256, 0, stream>>>(gates, hbuf, amax, t, outputs);
  }

  nll_kernel<<<B_SZ * T_STEPS, 256, 0, stream>>>(outputs, target, nll);
  loss_kernel<<<1, 256, 0, stream>>>(nll, loss);
}